// ReconLoss_33663953666360
// MI455X (gfx1250) — compile-verified
//
#include <hip/hip_runtime.h>
#include <math.h>

// CDNA5 / gfx1250, wave32. f32 WMMA path: V_WMMA_F32_16X16X4_F32.
typedef __attribute__((ext_vector_type(2))) float v2f;
typedef __attribute__((ext_vector_type(8))) float v8f;

#define WAVES_PER_BLOCK 8
#define TPB (WAVES_PER_BLOCK * 32)
#define STRIPS_PER_WAVE 2                       // 16-row strips per wave
#define ROWS_PER_WAVE (STRIPS_PER_WAVE * 16)    // 32
#define ROWS_PER_BLOCK (WAVES_PER_BLOCK * ROWS_PER_WAVE)  // 256

// Raw f32 mins: inputs here are always finite (squared distances), so skip
// fminf's IEEE sNaN canonicalization (v_max_num v,v,v) that hipcc inserts.
__device__ __forceinline__ float fmin_raw(float a, float b) {
  float r;
  asm("v_min_num_f32 %0, %1, %2" : "=v"(r) : "v"(a), "v"(b));
  return r;
}
__device__ __forceinline__ float fmin3_raw(float a, float b, float c) {
  float r;
  asm("v_min3_num_f32 %0, %1, %2, %3" : "=v"(r) : "v"(a), "v"(b), "v"(c));
  return r;
}

// Row-min of pairwise squared distances: out[b][n] = min_m ||X[b,n]-Y[b,m]||^2.
// Each wave owns STRIPS_PER_WAVE 16-row strips of X (loop-invariant A operands,
// 2 VGPRs each). Column loop is hand-unrolled by two 16-col Y tiles; each
// (strip, tile) pair is one v_wmma_f32_16x16x4_f32:
//   A[m]   = (x0, x1, x2, |x|^2)   (16x4: lanes 0-15 K={0,1}, 16-31 K={2,3})
//   B[:,n] = (-2y0, -2y1, -2y2, 1)
//   C[m][n]= |y_n|^2               (splat per lane; C/D col = lane%16)
// => D = |x|^2 - 2x.y + |y|^2 exactly, f32 RNE (matches reference precision).
// Accumulate with v_min3_num (acc, D_tile0, D_tile1).
__global__ __launch_bounds__(TPB) void chamfer_rowmin_kernel(
    const float* __restrict__ X, const float* __restrict__ Y,
    float* __restrict__ outmin, int N, int M)
{
  const int lane = threadIdx.x & 31;
  const int wave = threadIdx.x >> 5;
  const int half = lane >> 4;     // 0: lanes 0-15, 1: lanes 16-31
  const int mrow = lane & 15;
  const int b    = blockIdx.y;
  const int rowBase = blockIdx.x * ROWS_PER_BLOCK + wave * ROWS_PER_WAVE;

  // Loop-invariant A operands, one per strip.
  v2f A[STRIPS_PER_WAVE];
  #pragma unroll
  for (int r = 0; r < STRIPS_PER_WAVE; ++r) {
    const float* xp = X + ((size_t)b * N + rowBase + r * 16 + mrow) * 3;
    const float x0 = xp[0], x1 = xp[1], x2 = xp[2];
    const float sx = x0 * x0 + x1 * x1 + x2 * x2;
    A[r].x = half ? x2 : x0;   // K=2 : K=0
    A[r].y = half ? sx : x1;   // K=3 : K=1
  }

  v8f minD[STRIPS_PER_WAVE];
  #pragma unroll
  for (int r = 0; r < STRIPS_PER_WAVE; ++r)
    #pragma unroll
    for (int v = 0; v < 8; ++v) minD[r][v] = 3.402823466e38f;

  const float* Yb = Y + (size_t)b * M * 3;
  for (int t = 0; t < M; t += 32) {
    // Two 16-col Y tiles per iteration.
    const float* yp0 = Yb + (size_t)(t + mrow) * 3;
    const float* yp1 = Yb + (size_t)(t + 16 + mrow) * 3;
    const float a0 = yp0[0], a1 = yp0[1], a2 = yp0[2];
    const float c0 = yp1[0], c1 = yp1[1], c2 = yp1[2];
    const float ty0 = a0 * a0 + a1 * a1 + a2 * a2;
    const float ty1 = c0 * c0 + c1 * c1 + c2 * c2;

    v2f B0, B1;
    B0.x = half ? (-2.0f * a2) : (-2.0f * a0);  // K=2 : K=0
    B0.y = half ? 1.0f         : (-2.0f * a1);  // K=3 : K=1
    B1.x = half ? (-2.0f * c2) : (-2.0f * c0);
    B1.y = half ? 1.0f         : (-2.0f * c1);

    v8f C0, C1;
    #pragma unroll
    for (int v = 0; v < 8; ++v) { C0[v] = ty0; C1[v] = ty1; }

    #pragma unroll
    for (int r = 0; r < STRIPS_PER_WAVE; ++r) {
      v8f D0 = __builtin_amdgcn_wmma_f32_16x16x4_f32(
          false, A[r], false, B0, (short)0, C0, false, false);
      v8f D1 = __builtin_amdgcn_wmma_f32_16x16x4_f32(
          false, A[r], false, B1, (short)0, C1, false, false);
      #pragma unroll
      for (int v = 0; v < 8; ++v)
        minD[r][v] = fmin3_raw(minD[r][v], D0[v], D1[v]);
    }
  }

  // Min across the 16 lanes of each half (xor masks 1..8 stay inside each
  // 16-lane group on wave32). Lanes 0-15 hold rows v, lanes 16-31 rows v+8.
  #pragma unroll
  for (int m = 1; m <= 8; m <<= 1)
    #pragma unroll
    for (int r = 0; r < STRIPS_PER_WAVE; ++r)
      #pragma unroll
      for (int v = 0; v < 8; ++v)
        minD[r][v] = fmin_raw(minD[r][v], __shfl_xor(minD[r][v], m, 32));

  if (mrow == 0) {  // lane 0 writes rows 0..7 of each strip, lane 16 rows 8..15
    #pragma unroll
    for (int r = 0; r < STRIPS_PER_WAVE; ++r) {
      float* op = outmin + (size_t)b * N + rowBase + r * 16 + half * 8;
      #pragma unroll
      for (int v = 0; v < 8; ++v) op[v] = minD[r][v];
    }
  }
}

// Single-block deterministic finalize: coarse L2 mean + fine chamfer mean.
__global__ __launch_bounds__(256) void finalize_kernel(
    const float* __restrict__ rc, const float* __restrict__ gc,
    const float* __restrict__ rowmin, const float* __restrict__ colmin,
    float* __restrict__ out, int nCoarsePts, int nFinePts)
{
  const int tid = threadIdx.x;
  float accC = 0.0f, accF = 0.0f;

  for (int i = tid; i < nCoarsePts; i += 256) {
    const float dx = rc[3 * i + 0] - gc[3 * i + 0];
    const float dy = rc[3 * i + 1] - gc[3 * i + 1];
    const float dz = rc[3 * i + 2] - gc[3 * i + 2];
    accC += sqrtf(dx * dx + dy * dy + dz * dz);
  }
  for (int i = tid; i < nFinePts; i += 256)
    accF += sqrtf(fmaxf(rowmin[i], 0.0f));
  for (int i = tid; i < nFinePts; i += 256)
    accF += sqrtf(fmaxf(colmin[i], 0.0f));

  __shared__ float sC[256];
  __shared__ float sF[256];
  sC[tid] = accC;
  sF[tid] = accF;
  __syncthreads();
  for (int s = 128; s > 0; s >>= 1) {
    if (tid < s) {
      sC[tid] += sC[tid + s];
      sF[tid] += sF[tid + s];
    }
    __syncthreads();
  }
  if (tid == 0) {
    out[0] = sC[0] / (float)nCoarsePts;                 // loss_coarse
    out[1] = 0.5f * sF[0] / (float)nFinePts;            // loss_fine
  }
}

extern "C" void kernel_launch(void* const* d_in, const int* in_sizes, int n_in,
                              void* d_out, int out_size, void* d_ws, size_t ws_size,
                              hipStream_t stream) {
  (void)n_in; (void)out_size; (void)ws_size;
  const float* ret_coarse = (const float*)d_in[0];  // [B, Nc, 3]
  const float* ret_fine   = (const float*)d_in[1];  // [B, Nf, 3]
  const float* gt_fine    = (const float*)d_in[2];  // [B, Nf, 3]
  const float* gt_coarse  = (const float*)d_in[3];  // [B, Nc, 3]
  float* out = (float*)d_out;

  const int B  = 4;
  const int Nc = in_sizes[0] / (B * 3);   // 1024
  const int Nf = in_sizes[1] / (B * 3);   // 8192

  float* rowmin = (float*)d_ws;           // B*Nf floats
  float* colmin = rowmin + (size_t)B * Nf;

  // 256 rows per block (8 waves x 2 strips x 16 rows).
  dim3 grid(Nf / ROWS_PER_BLOCK, B);
  chamfer_rowmin_kernel<<<grid, TPB, 0, stream>>>(ret_fine, gt_fine, rowmin, Nf, Nf);
  chamfer_rowmin_kernel<<<grid, TPB, 0, stream>>>(gt_fine, ret_fine, colmin, Nf, Nf);
  finalize_kernel<<<1, 256, 0, stream>>>(ret_coarse, gt_coarse, rowmin, colmin,
                                         out, B * Nc, B * Nf);
}